// KARNN_86921548137052
// MI455X (gfx1250) — compile-verified
//
#include <hip/hip_runtime.h>
#include <math.h>

// ---------------------------------------------------------------------------
// KAN-RNN for MI455X (gfx1250, wave32).
//  Fold: y[b,o] = sum_i sb[i,o]*silu(x) + ss[i,o]*sum_n B_n(x)*coef[i,o,n]
//      = phi(x) @ W, phi = per-input [silu, B0..B7] (9 features).
//  Kernel 1: persistent grid-synced fp32 GEMV recurrence (W_h L2-resident,
//            K-quad packed -> b128 weight loads).
//  Kernel 2: fp32 WMMA GEMM (V_WMMA_F32_16X16X4_F32), W_o K-pair packed
//            -> b64 B-fragment loads.
// ---------------------------------------------------------------------------

typedef float v2f __attribute__((ext_vector_type(2)));
typedef float v4f __attribute__((ext_vector_type(4)));
typedef float v8f __attribute__((ext_vector_type(8)));

#define T_STEPS 4096
#define IN_DIM  512
#define HID     1024
#define OUT_DIM 512
#define NCOEF   8            // num + k
#define NKNOT   12           // num + 2k + 1
#define INH     (IN_DIM + HID)      // 1536
#define KH      (INH * 9)           // 13824 (div by 4)
#define KO      (HID * 9)           // 9216  (div by 2)
#define NBLK_RNN 64

#if defined(__has_builtin)
# if __has_builtin(__builtin_amdgcn_wmma_f32_16x16x4_f32)
#  define HAVE_WMMA_F32X4 1
# endif
#endif
#ifndef HAVE_WMMA_F32X4
# define HAVE_WMMA_F32X4 0
#endif

__device__ __forceinline__ float silu_f(float v) {
    return v / (1.0f + expf(-v));
}

// Cox-de Boor cubic B-spline basis, 12 knots -> 8 basis values.
__device__ __forceinline__ void bspline8(float x, const float* __restrict__ g,
                                         float* __restrict__ out) {
    float B[11];
#pragma unroll
    for (int j = 0; j < 11; ++j)
        B[j] = (x >= g[j] && x < g[j + 1]) ? 1.0f : 0.0f;
#pragma unroll
    for (int d = 1; d <= 3; ++d) {
#pragma unroll 11
        for (int j = 0; j < 11 - d; ++j) {
            float left  = (x - g[j]) / (g[j + d] - g[j]);
            float right = (g[j + d + 1] - x) / (g[j + d + 1] - g[j + 1]);
            B[j] = left * B[j] + right * B[j + 1];
        }
    }
#pragma unroll
    for (int j = 0; j < 8; ++j) out[j] = B[j];
}

// ---------------------------------------------------------------------------
// Pack folded weights, K-QUAD interleaved: W4[(k>>2)][o][k&3]  (for GEMV b128)
// ---------------------------------------------------------------------------
__global__ void pack_w_quad_kernel(const float* __restrict__ sb,
                                   const float* __restrict__ ss,
                                   const float* __restrict__ coef,
                                   float* __restrict__ W, int in_dim, int out_dim) {
    int idx = blockIdx.x * blockDim.x + threadIdx.x;
    int total = in_dim * out_dim;
    if (idx >= total) return;
    int i = idx / out_dim;
    int o = idx - i * out_dim;
    float s_b = sb[idx];
    float s_s = ss[idx];
    const float* c = coef + (size_t)idx * NCOEF;
#pragma unroll
    for (int ccomp = 0; ccomp < 9; ++ccomp) {
        int k = i * 9 + ccomp;
        float v = (ccomp == 0) ? s_b : s_s * c[ccomp - 1];
        W[((size_t)(k >> 2) * out_dim + o) * 4 + (k & 3)] = v;
    }
}

// ---------------------------------------------------------------------------
// Pack folded weights, K-PAIR interleaved: W2[(k>>1)][o][k&1]  (for WMMA b64)
// ---------------------------------------------------------------------------
__global__ void pack_w_pair_kernel(const float* __restrict__ sb,
                                   const float* __restrict__ ss,
                                   const float* __restrict__ coef,
                                   float* __restrict__ W, int in_dim, int out_dim) {
    int idx = blockIdx.x * blockDim.x + threadIdx.x;
    int total = in_dim * out_dim;
    if (idx >= total) return;
    int i = idx / out_dim;
    int o = idx - i * out_dim;
    float s_b = sb[idx];
    float s_s = ss[idx];
    const float* c = coef + (size_t)idx * NCOEF;
#pragma unroll
    for (int ccomp = 0; ccomp < 9; ++ccomp) {
        int k = i * 9 + ccomp;
        float v = (ccomp == 0) ? s_b : s_s * c[ccomp - 1];
        W[((size_t)(k >> 1) * out_dim + o) * 2 + (k & 1)] = v;
    }
}

// ---------------------------------------------------------------------------
// Kernel 1: persistent recurrence. 64 blocks x 256 threads, 16 outputs/block.
// Thread (kslice, oi): accumulates k-quads j = kslice, kslice+16, ... (216 it)
//   phi quad   : ds_load_b128 (broadcast within half-wave)
//   weight quad: global_load_b128, 16 lanes -> 256B contiguous
// Grid barrier: monotonic atomic counter (zeroed by captured memset node).
// ---------------------------------------------------------------------------
__global__ __launch_bounds__(256) void kan_rnn_kernel(
    const float* __restrict__ x,       // (T, IN_DIM)
    const float* __restrict__ h0,      // (HID)
    const float* __restrict__ grid_h,  // (INH, 12) rows identical
    const float* __restrict__ Wh4,     // quad-packed (KH/4, HID, 4)
    float* __restrict__ hbuf,          // (2, HID)
    unsigned* __restrict__ bar,        // barrier counter, zero at launch
    float* __restrict__ hidden_out)    // (T, HID)
{
    __shared__ __align__(16) float phi[KH];   // 55296 B
    __shared__ float red[256];
    __shared__ float knots[NKNOT];

    const int tid = threadIdx.x;
    if (tid < NKNOT) knots[tid] = grid_h[tid];

    const int oi     = tid & 15;
    const int kslice = tid >> 4;              // 0..15
    const int o      = blockIdx.x * 16 + oi;

    for (int t = 0; t < T_STEPS; ++t) {
        const float* hprev = (t == 0) ? h0 : &hbuf[(t & 1) * HID];
        __syncthreads();

        // ---- build phi[1536*9] in LDS (6 inputs per thread) ----
#pragma unroll
        for (int u = 0; u < INH / 256; ++u) {
            int i = tid + u * 256;
            float v = (i < IN_DIM) ? x[(size_t)t * IN_DIM + i] : hprev[i - IN_DIM];
            float B[8];
            bspline8(v, knots, B);
            float* dst = &phi[i * 9];
            dst[0] = silu_f(v);
#pragma unroll
            for (int n = 0; n < 8; ++n) dst[1 + n] = B[n];
        }
        __syncthreads();

        // ---- GEMV over k-quads ----
        v4f acc = {0.f, 0.f, 0.f, 0.f};
#pragma unroll 4
        for (int j = kslice; j < KH / 4; j += 16) {
            const v4f ph = *(const v4f*)&phi[j * 4];
            const v4f w  = *(const v4f*)&Wh4[((size_t)j * HID + o) * 4];
            acc.x = fmaf(ph.x, w.x, acc.x);
            acc.y = fmaf(ph.y, w.y, acc.y);
            acc.z = fmaf(ph.z, w.z, acc.z);
            acc.w = fmaf(ph.w, w.w, acc.w);
        }
        red[tid] = (acc.x + acc.y) + (acc.z + acc.w);
        __syncthreads();
#pragma unroll
        for (int s = 128; s >= 16; s >>= 1) {
            if (tid < s) red[tid] += red[tid + s];
            __syncthreads();
        }
        if (tid < 16) {
            float hv = red[tid];
            int oo = blockIdx.x * 16 + tid;
            hbuf[((t + 1) & 1) * HID + oo] = hv;
            hidden_out[(size_t)t * HID + oo] = hv;
        }
        __syncthreads();

        // ---- grid-wide barrier ----
        if (tid == 0) {
            __threadfence();
            atomicAdd(bar, 1u);
            unsigned target = (unsigned)(t + 1) * NBLK_RNN;
            while (__hip_atomic_load(bar, __ATOMIC_ACQUIRE,
                                     __HIP_MEMORY_SCOPE_AGENT) < target)
                __builtin_amdgcn_s_sleep(2);
        }
        __syncthreads();
    }
}

// ---------------------------------------------------------------------------
// Kernel 2: output-layer GEMM via fp32 WMMA (16x16x4).
// C tile 64x128 per block; 8 waves; each wave: 4 accumulators of 16x16.
// A (phi of hidden) built on the fly in LDS, 8 inputs (72 K-values) at a time.
// f32 WMMA fragment layouts (ISA §7.12.2):
//   A (16x4):  lane<16 -> M=lane,   {K=k0,k0+1}; lane>=16 -> M=lane-16, {K=k0+2,k0+3}
//   B (4x16):  lane<16 -> N=lane,   {K=k0,k0+1}; lane>=16 -> N=lane-16, {K=k0+2,k0+3}
//   C/D (v8f): reg r, lane<16 -> C[r][lane]; lane>=16 -> C[8+r][lane-16]
// ---------------------------------------------------------------------------
#define TM 64
#define TN 128
#define APAD 74   // even pad -> 8B-aligned v2f LDS loads; near-conflict-free

__global__ __launch_bounds__(256) void kan_out_gemm_kernel(
    const float* __restrict__ hidden,  // (T, HID)
    const float* __restrict__ grid_o,  // (HID, 12)
    const float* __restrict__ WoP,     // pair-packed (KO/2, OUT_DIM, 2)
    float* __restrict__ out)           // (T, OUT_DIM)
{
    __shared__ __align__(16) float Alds[TM * APAD];   // 18944 B
    __shared__ float knots[NKNOT];

    const int tid = threadIdx.x;
    if (tid < NKNOT) knots[tid] = grid_o[tid];
    __syncthreads();

    const int m0 = blockIdx.x * TM;
    const int n0 = blockIdx.y * TN;

    const int wave = tid >> 5;
    const int lane = tid & 31;
    const int half = lane >> 4;     // 0|1
    const int l16  = lane & 15;
    const int mi    = wave >> 1;          // 0..3 -> M sub-tile row
    const int nbase = (wave & 1) * 64;    // 0|64 -> N half

    const int r = tid >> 2;  // 0..63 : row for phi staging
    const int q = tid & 3;   // 0..3  : input-pair for phi staging

#if HAVE_WMMA_F32X4
    v8f zero = {};
    v8f acc[4] = {zero, zero, zero, zero};
#else
    float acc_s[32];
#pragma unroll
    for (int i = 0; i < 32; ++i) acc_s[i] = 0.f;
    const int s_col  = tid & 127;
    const int s_mrow = (tid >> 7) * 32;
#endif

    for (int ic = 0; ic < HID / 8; ++ic) {     // 128 chunks of 8 inputs
        __syncthreads();                        // guard previous consumption
        // ---- stage phi chunk: rows m0..m0+63, inputs ic*8..ic*8+7 ----
#pragma unroll
        for (int u = 0; u < 2; ++u) {
            int ii = q * 2 + u;                 // 0..7
            int i  = ic * 8 + ii;
            float v = hidden[(size_t)(m0 + r) * HID + i];
            float B[8];
            bspline8(v, knots, B);
            float* dst = &Alds[r * APAD + ii * 9];
            dst[0] = silu_f(v);
#pragma unroll
            for (int n = 0; n < 8; ++n) dst[1 + n] = B[n];
        }
        __syncthreads();

#if HAVE_WMMA_F32X4
        // ---- consume 72 K-values as 18 WMMA steps of K=4 ----
#pragma unroll 2
        for (int kk = 0; kk < 18; ++kk) {
            const int acol = kk * 4 + half * 2;            // even
            const v2f a = *(const v2f*)&Alds[(mi * 16 + l16) * APAD + acol];
            const int k2 = (ic * 72 + kk * 4 + half * 2) >> 1;   // pair index
#pragma unroll
            for (int ns = 0; ns < 4; ++ns) {
                const int ncol = n0 + nbase + ns * 16 + l16;
                const v2f b = *(const v2f*)&WoP[((size_t)k2 * OUT_DIM + ncol) * 2];
                acc[ns] = __builtin_amdgcn_wmma_f32_16x16x4_f32(
                    false, a, false, b, (short)0, acc[ns], false, false);
            }
        }
#else
        // ---- scalar fallback (pair-packed Wo) ----
        for (int kk = 0; kk < 72; ++kk) {
            int k = ic * 72 + kk;
            float w = WoP[((size_t)(k >> 1) * OUT_DIM + n0 + s_col) * 2 + (k & 1)];
#pragma unroll
            for (int rr = 0; rr < 32; ++rr)
                acc_s[rr] = fmaf(Alds[(s_mrow + rr) * APAD + kk], w, acc_s[rr]);
        }
#endif
    }

#if HAVE_WMMA_F32X4
#pragma unroll
    for (int ns = 0; ns < 4; ++ns) {
#pragma unroll
        for (int rr = 0; rr < 8; ++rr) {
            int row = m0 + mi * 16 + half * 8 + rr;
            int col = n0 + nbase + ns * 16 + l16;
            out[(size_t)row * OUT_DIM + col] = acc[ns][rr];
        }
    }
#else
#pragma unroll
    for (int rr = 0; rr < 32; ++rr)
        out[(size_t)(m0 + s_mrow + rr) * OUT_DIM + n0 + s_col] = acc_s[rr];
#endif
}

// ---------------------------------------------------------------------------
// Host wrapper
// ---------------------------------------------------------------------------
extern "C" void kernel_launch(void* const* d_in, const int* in_sizes, int n_in,
                              void* d_out, int out_size, void* d_ws, size_t ws_size,
                              hipStream_t stream) {
    const float* x      = (const float*)d_in[0];
    const float* h0     = (const float*)d_in[1];
    const float* grid_h = (const float*)d_in[2];
    const float* coef_h = (const float*)d_in[3];
    const float* sb_h   = (const float*)d_in[4];
    const float* ss_h   = (const float*)d_in[5];
    const float* grid_o = (const float*)d_in[6];
    const float* coef_o = (const float*)d_in[7];
    const float* sb_o   = (const float*)d_in[8];
    const float* ss_o   = (const float*)d_in[9];

    float* out_f   = (float*)d_out;
    float* outputs = out_f;                              // (T, OUT_DIM)
    float* hidden  = out_f + (size_t)T_STEPS * OUT_DIM;  // (T, HID)

    char* ws = (char*)d_ws;
    const size_t offWh   = 0;                                        // 56,623,104 B
    const size_t offWo   = offWh + (size_t)KH * HID * sizeof(float); // 18,874,368 B
    const size_t offHbuf = offWo + (size_t)KO * OUT_DIM * sizeof(float);
    const size_t offBar  = offHbuf + 2 * HID * sizeof(float);

    float*    Wh4  = (float*)(ws + offWh);
    float*    WoP  = (float*)(ws + offWo);
    float*    hbuf = (float*)(ws + offHbuf);
    unsigned* bar  = (unsigned*)(ws + offBar);

    // zero the grid-barrier counter (capturable as a memset node)
    hipMemsetAsync(bar, 0, 64, stream);

    // fold + repack weights
    {
        int total = INH * HID;
        pack_w_quad_kernel<<<(total + 255) / 256, 256, 0, stream>>>(
            sb_h, ss_h, coef_h, Wh4, INH, HID);
    }
    {
        int total = HID * OUT_DIM;
        pack_w_pair_kernel<<<(total + 255) / 256, 256, 0, stream>>>(
            sb_o, ss_o, coef_o, WoP, HID, OUT_DIM);
    }

    // recurrence (persistent, grid-synced)
    kan_rnn_kernel<<<NBLK_RNN, 256, 0, stream>>>(
        x, h0, grid_h, Wh4, hbuf, bar, hidden);

    // output layer GEMM via fp32 WMMA
    dim3 g2(T_STEPS / TM, OUT_DIM / TN);
    kan_out_gemm_kernel<<<g2, 256, 0, stream>>>(hidden, grid_o, WoP, outputs);
}